// GINDrugEncoder_1812476199544
// MI455X (gfx1250) — compile-verified
//
#include <hip/hip_runtime.h>

#define NG 1024
#define H 128
#define F_IN 78
#define F_PAD 80
#define BN_EPS 1e-5f

typedef float v2f __attribute__((ext_vector_type(2)));
typedef float v8f __attribute__((ext_vector_type(8)));

static __device__ __forceinline__ void atomAddF(float* p, float v) {
    unsafeAtomicAdd(p, v);   // -> global_atomic_add_f32
}

// ---------------------------------------------------------------- utilities
__global__ void fill_f32(float* __restrict__ p, float v, long n) {
    long i = (long)blockIdx.x * blockDim.x + threadIdx.x;
    if (i < n) p[i] = v;
}
__global__ void fill_u32(unsigned* __restrict__ p, unsigned v, long n) {
    long i = (long)blockIdx.x * blockDim.x + threadIdx.x;
    if (i < n) p[i] = v;
}

// copy x [N,78] into padded [N,80] (cols 78,79 = 0)
__global__ void pad_copy_x(const float* __restrict__ x, float* __restrict__ hp, long N) {
    long i = (long)blockIdx.x * blockDim.x + threadIdx.x;
    long total = N * F_PAD;
    if (i >= total) return;
    long n = i / F_PAD;
    int  c = (int)(i - n * F_PAD);
    hp[i] = (c < F_IN) ? x[n * F_IN + c] : 0.0f;
}

// pad W1_0 [78,128] -> [80,128] with zero rows
__global__ void pad_w1(const float* __restrict__ w, float* __restrict__ wp) {
    int i = blockIdx.x * blockDim.x + threadIdx.x;
    if (i >= F_PAD * H) return;
    int r = i / H;
    wp[i] = (r < F_IN) ? w[i] : 0.0f;
}

// ------------------------------------------------- edge scatter-add (GIN agg)
// hp[dst] += xs[src]; 2D grid: x over edges, y over feature chunks.
template <int CHUNK>
__global__ void edge_agg(const int* __restrict__ src, const int* __restrict__ dst,
                         const float* __restrict__ xs, float* __restrict__ hp,
                         int nE, int sS, int sD) {
    int e = blockIdx.x * blockDim.x + threadIdx.x;
    if (e >= nE) return;
    int c = blockIdx.y * CHUNK;
    int s = src[e], d = dst[e];
    long so = (long)s * sS + c;
    long dofs = (long)d * sD + c;
#pragma unroll
    for (int j = 0; j < CHUNK; j++) atomAddF(&hp[dofs + j], xs[so + j]);
}

// --------------------------------------------- fused 2-layer MLP (fp32 WMMA)
// h2 = relu(relu(hpre @ W1 + b1) @ W2 + b2); also accumulates per-col sum/sumsq.
// 256 threads = 8 waves; 32 rows per block (N % 32 == 0 -> EXEC always all-1s).
template <int K1>
__global__ __launch_bounds__(256) void fused_mlp(
    const float* __restrict__ hpre,  // [N, K1]
    const float* __restrict__ W1,    // [K1, H]
    const float* __restrict__ b1,    // [H]
    const float* __restrict__ W2,    // [H, H]
    const float* __restrict__ b2,    // [H]
    float* __restrict__ hout,        // [N, H]
    float* __restrict__ stats)       // [2*H]: colsum, colsumsq
{
    constexpr int ROWS = 32;
    constexpr int PAD  = 4;  // LDS bank-conflict padding (stride%64 = 4)
    __shared__ float At[ROWS][K1 + PAD];
    __shared__ float Ht[ROWS][H + PAD];

    const int tid  = threadIdx.x;
    const int wave = tid >> 5;      // 0..7 -> 16-col tile
    const int lane = tid & 31;
    const int l16  = lane & 15;
    const int hi   = lane >> 4;     // lane half: selects K pair (A/B) / row octet (C)
    const long rowbase = (long)blockIdx.x * ROWS;
    const int  col = wave * 16 + l16;

    // stage activation tile
    for (int t = tid; t < ROWS * K1; t += 256) {
        int r = t / K1, c = t - r * K1;
        At[r][c] = hpre[(rowbase + r) * K1 + c];
    }
    __syncthreads();

    // ---------------- GEMM1: [32 x K1] @ [K1 x 128]
    v8f c0, c1;
    {
        float bb = b1[col];
#pragma unroll
        for (int v = 0; v < 8; v++) { c0[v] = bb; c1[v] = bb; }
    }
#pragma unroll 4
    for (int kk = 0; kk < K1 / 4; kk++) {
        const int k0 = kk * 4 + hi * 2;
        v2f b;  b.x  = W1[(k0 + 0) * H + col]; b.y  = W1[(k0 + 1) * H + col];
        v2f a0; a0.x = At[l16][k0];            a0.y = At[l16][k0 + 1];
        v2f a1; a1.x = At[16 + l16][k0];       a1.y = At[16 + l16][k0 + 1];
        c0 = __builtin_amdgcn_wmma_f32_16x16x4_f32(false, a0, false, b, (short)0, c0, false, false);
        c1 = __builtin_amdgcn_wmma_f32_16x16x4_f32(false, a1, false, b, (short)0, c1, false, false);
    }
#pragma unroll
    for (int v = 0; v < 8; v++) { c0[v] = fmaxf(c0[v], 0.f); c1[v] = fmaxf(c1[v], 0.f); }

    // scatter h1 tile to LDS (C layout: lane half -> row octet)
#pragma unroll
    for (int v = 0; v < 8; v++) {
        Ht[v + hi * 8][col]      = c0[v];
        Ht[16 + v + hi * 8][col] = c1[v];
    }
    __syncthreads();

    // ---------------- GEMM2: [32 x 128] @ [128 x 128]
    {
        float bb = b2[col];
#pragma unroll
        for (int v = 0; v < 8; v++) { c0[v] = bb; c1[v] = bb; }
    }
#pragma unroll 4
    for (int kk = 0; kk < H / 4; kk++) {
        const int k0 = kk * 4 + hi * 2;
        v2f b;  b.x  = W2[(k0 + 0) * H + col]; b.y  = W2[(k0 + 1) * H + col];
        v2f a0; a0.x = Ht[l16][k0];            a0.y = Ht[l16][k0 + 1];
        v2f a1; a1.x = Ht[16 + l16][k0];       a1.y = Ht[16 + l16][k0 + 1];
        c0 = __builtin_amdgcn_wmma_f32_16x16x4_f32(false, a0, false, b, (short)0, c0, false, false);
        c1 = __builtin_amdgcn_wmma_f32_16x16x4_f32(false, a1, false, b, (short)0, c1, false, false);
    }

    float s = 0.f, sq = 0.f;
#pragma unroll
    for (int v = 0; v < 8; v++) {
        c0[v] = fmaxf(c0[v], 0.f);
        c1[v] = fmaxf(c1[v], 0.f);
        s  += c0[v] + c1[v];
        sq += c0[v] * c0[v] + c1[v] * c1[v];
        hout[(rowbase + v + hi * 8) * H + col]      = c0[v];
        hout[(rowbase + 16 + v + hi * 8) * H + col] = c1[v];
    }
    // combine the two row-octet halves that share a column (wave32 xor-16)
    s  += __shfl_xor(s, 16);
    sq += __shfl_xor(sq, 16);
    if (hi == 0) {
        atomAddF(&stats[col], s);
        atomAddF(&stats[H + col], sq);
    }
}

// ---------------------------------------------------------------- batchnorm
__global__ void bn_finalize(const float* __restrict__ stats,
                            const float* __restrict__ gamma, const float* __restrict__ beta,
                            float* __restrict__ scsh, float invN) {
    int c = threadIdx.x;  // H threads
    float mu  = stats[c] * invN;
    float var = stats[H + c] * invN - mu * mu;
    float sc  = gamma[c] / sqrtf(var + BN_EPS);
    scsh[c]     = sc;
    scsh[H + c] = beta[c] - mu * sc;
}

__global__ void bn_apply(const float* __restrict__ hm, const float* __restrict__ scsh,
                         const float* __restrict__ resid, float* __restrict__ out, long total) {
    long i = (long)blockIdx.x * blockDim.x + threadIdx.x;
    if (i >= total) return;
    int c = (int)(i & (H - 1));
    float v = hm[i] * scsh[c] + scsh[H + c];
    if (resid) v += resid[i];
    out[i] = v;
}

// ----------------------------------------------------------------- pooling
__global__ void pool_kernel(const float* __restrict__ h, const int* __restrict__ batch,
                            float* __restrict__ psum, int* __restrict__ pmax,
                            int* __restrict__ pcnt, long N) {
    long i = (long)blockIdx.x * blockDim.x + threadIdx.x;
    if (i >= N * H) return;
    long n = i >> 7;
    int  c = (int)(i & (H - 1));
    int  g = batch[n];
    float v = h[i];
    atomAddF(&psum[(long)g * H + c], v);
    int* addr = &pmax[(long)g * H + c];
    if (v >= 0.f) atomicMax(addr, __float_as_int(v));
    else          atomicMin((unsigned int*)addr, __float_as_uint(v));
    if (c == 0) atomicAdd(&pcnt[g], 1);
}

__global__ void final_gemm(const float* __restrict__ psum, const int* __restrict__ pmax,
                           const int* __restrict__ pcnt, const float* __restrict__ Wf,
                           const float* __restrict__ bf, float* __restrict__ out) {
    int i = blockIdx.x * blockDim.x + threadIdx.x;  // NG*H threads
    int g = i >> 7, j = i & (H - 1);
    int cnt = pcnt[g];
    float inv = cnt > 0 ? 1.f / (float)cnt : 0.f;
    float acc = bf[j];
#pragma unroll 4
    for (int c = 0; c < H; c++) {
        float mean = psum[(long)g * H + c] * inv;
        float mx   = cnt > 0 ? __int_as_float(pmax[(long)g * H + c]) : 0.f;
        acc += mean * Wf[c * H + j] + mx * Wf[(H + c) * H + j];
    }
    out[i] = acc;
}

// ---------------------------------------------------------------- launcher
static inline long cdivl(long a, long b) { return (a + b - 1) / b; }

extern "C" void kernel_launch(void* const* d_in, const int* in_sizes, int n_in,
                              void* d_out, int out_size, void* d_ws, size_t ws_size,
                              hipStream_t stream) {
    const float* x     = (const float*)d_in[0];
    const int*   ei    = (const int*)d_in[1];
    const int*   batch = (const int*)d_in[2];
    const float* W1_0  = (const float*)d_in[3];
    const float* b1_0  = (const float*)d_in[4];
    const float* W2_0  = (const float*)d_in[5];
    const float* b2_0  = (const float*)d_in[6];
    const float* W1    = (const float*)d_in[7];
    const float* b1    = (const float*)d_in[8];
    const float* W2    = (const float*)d_in[9];
    const float* b2    = (const float*)d_in[10];
    const float* gamma = (const float*)d_in[11];
    const float* beta  = (const float*)d_in[12];
    const float* Wf    = (const float*)d_in[13];
    const float* bf    = (const float*)d_in[14];

    const long N = in_sizes[0] / F_IN;   // 100000 (multiple of 32)
    const int  E = in_sizes[1] / 2;      // 3200000
    const int* src = ei;
    const int* dst = ei + E;

    // workspace carve-up
    float* bufA  = (float*)d_ws;                 // [N,H] current activations
    float* bufB  = bufA + (size_t)N * H;         // [N,H] hpre / next activations
    float* bufC  = bufB + (size_t)N * H;         // [N,H] MLP output
    float* Wpad  = bufC + (size_t)N * H;         // [80,128]
    float* stats = Wpad + F_PAD * H;             // [2H]
    float* scsh  = stats + 2 * H;                // [2H]
    float* psum  = scsh + 2 * H;                 // [NG,H]
    int*   pmax  = (int*)(psum + (size_t)NG * H);
    int*   pcnt  = pmax + (size_t)NG * H;

    const long totNH = N * H;
    const dim3 blk(256);

    // ---------------- layer 0 (78 -> 128, padded K=80)
    pad_copy_x<<<(unsigned)cdivl(N * F_PAD, 256), blk, 0, stream>>>(x, bufB, N);
    edge_agg<2><<<dim3((unsigned)cdivl(E, 256), F_IN / 2), blk, 0, stream>>>(
        src, dst, x, bufB, E, F_IN, F_PAD);
    pad_w1<<<(F_PAD * H + 255) / 256, blk, 0, stream>>>(W1_0, Wpad);
    fill_f32<<<1, blk, 0, stream>>>(stats, 0.f, 2 * H);
    fused_mlp<F_PAD><<<(unsigned)(N / 32), blk, 0, stream>>>(
        bufB, Wpad, b1_0, W2_0, b2_0, bufC, stats);
    bn_finalize<<<1, H, 0, stream>>>(stats, gamma, beta, scsh, 1.f / (float)N);
    bn_apply<<<(unsigned)cdivl(totNH, 256), blk, 0, stream>>>(
        bufC, scsh, (const float*)nullptr, bufA, totNH);

    // ---------------- layers 1..4 (residual)
    float* cur = bufA;
    float* pre = bufB;
    for (int i = 0; i < 4; i++) {
        hipMemcpyAsync(pre, cur, (size_t)totNH * sizeof(float),
                       hipMemcpyDeviceToDevice, stream);
        edge_agg<4><<<dim3((unsigned)cdivl(E, 256), H / 4), blk, 0, stream>>>(
            src, dst, cur, pre, E, H, H);
        fill_f32<<<1, blk, 0, stream>>>(stats, 0.f, 2 * H);
        fused_mlp<H><<<(unsigned)(N / 32), blk, 0, stream>>>(
            pre, W1 + (size_t)i * H * H, b1 + i * H,
            W2 + (size_t)i * H * H, b2 + i * H, bufC, stats);
        bn_finalize<<<1, H, 0, stream>>>(stats, gamma + (i + 1) * H, beta + (i + 1) * H,
                                         scsh, 1.f / (float)N);
        bn_apply<<<(unsigned)cdivl(totNH, 256), blk, 0, stream>>>(
            bufC, scsh, cur, pre, totNH);
        float* t = cur; cur = pre; pre = t;  // rotate: result is new cur
    }

    // ---------------- pooling + readout
    fill_f32<<<(NG * H + 255) / 256, blk, 0, stream>>>(psum, 0.f, NG * H);
    fill_u32<<<(NG * H + 255) / 256, blk, 0, stream>>>((unsigned*)pmax, 0xFF800000u, NG * H);
    fill_u32<<<(NG + 255) / 256, blk, 0, stream>>>((unsigned*)pcnt, 0u, NG);
    pool_kernel<<<(unsigned)cdivl(totNH, 256), blk, 0, stream>>>(
        cur, batch, psum, pmax, pcnt, N);
    final_gemm<<<(NG * H) / 256, blk, 0, stream>>>(psum, pmax, pcnt, Wf, bf, (float*)d_out);
}